// NMS_30408368456307
// MI455X (gfx1250) — compile-verified
//
#include <hip/hip_runtime.h>
#include <stdint.h>

#define NBOX 4096
#define NWORDS (NBOX / 32)          // 128 mask words per row
#define IOU_T 0.5f
#define CONF_T 0.25f

typedef __attribute__((ext_vector_type(2))) float v2f;
typedef __attribute__((ext_vector_type(8))) float v8f;

// Builtin signature (from hipcc diagnostic): AS1 is printed as '__device__',
// and the expected param is a non-const 'int*' in AS1; dst is AS3 ('__shared__').
typedef int __attribute__((address_space(1))) g_i32;
typedef int __attribute__((address_space(3))) l_i32;

#if defined(__has_builtin)
#if __has_builtin(__builtin_amdgcn_global_load_async_to_lds_b32)
#define USE_ASYNC_LDS 1
#define ASYNC_LOAD_B32(gsrc, ldst) \
    __builtin_amdgcn_global_load_async_to_lds_b32((g_i32*)(gsrc), (l_i32*)(ldst), 0, 0)
#if __has_builtin(__builtin_amdgcn_s_wait_asynccnt)
#define WAIT_ASYNC0() __builtin_amdgcn_s_wait_asynccnt(0)
#else
#define WAIT_ASYNC0() asm volatile("s_wait_asynccnt 0" ::: "memory")
#endif
#endif
#endif

// ---------------------------------------------------------------------------
// Kernel 1: bitonic argsort by descending confidence, entirely in LDS.
// One workgroup, 1024 threads (32 waves). key = -conf so ascending sort
// yields descending confidence. Emits sorted boxes (float4) + sorted scores.
// ---------------------------------------------------------------------------
__global__ void nms_sort_kernel(const float* __restrict__ xyxy,
                                const float* __restrict__ conf,
                                float4* __restrict__ sboxes,
                                float* __restrict__ sscores) {
    __shared__ float key[NBOX];
    __shared__ int   val[NBOX];
    const int tid = threadIdx.x;

    for (int i = tid; i < NBOX; i += 1024) { key[i] = -conf[i]; val[i] = i; }
    __syncthreads();

    for (int k = 2; k <= NBOX; k <<= 1) {
        for (int j = k >> 1; j > 0; j >>= 1) {
            for (int i = tid; i < NBOX; i += 1024) {
                int ixj = i ^ j;
                if (ixj > i) {
                    bool up = ((i & k) == 0);
                    float ki = key[i], kj = key[ixj];
                    if (up ? (ki > kj) : (ki < kj)) {
                        key[i] = kj; key[ixj] = ki;
                        int t = val[i]; val[i] = val[ixj]; val[ixj] = t;
                    }
                }
            }
            __syncthreads();
        }
    }

    const float4* xy4 = (const float4*)xyxy;
    for (int i = tid; i < NBOX; i += 1024) {
        sscores[i] = -key[i];
        sboxes[i]  = xy4[val[i]];
    }
}

// ---------------------------------------------------------------------------
// Kernel 2: suppression bitmask build. Each wave32 owns a 16-row x 32-col
// tile. The union's area[i]+area[j] broadcast term is computed on the matrix
// pipe via V_WMMA_F32_16X16X4_F32 as a rank-2 update (co-executes with the
// VALU min/max intersection chain). __ballot (wave32) packs the per-pair
// iou>thresh predicate: accumulator r -> row r bits[15:0], row r+8 bits[31:16].
// Two 16-col sub-tiles assemble one 32-bit mask word per row.
// ---------------------------------------------------------------------------
__global__ void nms_mask_kernel(const float4* __restrict__ sboxes,
                                uint32_t* __restrict__ rowmask) {
    const int lane   = threadIdx.x & 31;
    const int waveId = threadIdx.x >> 5;
    const int tileId = blockIdx.x * 8 + waveId;     // 32768 tiles total
    const int rowBase = (tileId >> 7) * 16;         // 256 row tiles
    const int colBase = (tileId & 127) * 32;        // 128 col tiles
    const bool hi = lane >= 16;
    const int  l16 = lane & 15;

    // Per-lane boxes for A-areas (rows) and the two column sub-tiles.
    float4 rbl = sboxes[rowBase + l16];
    float rowArea = (rbl.z - rbl.x) * (rbl.w - rbl.y);
    float4 cb0 = sboxes[colBase + l16];
    float4 cb1 = sboxes[colBase + 16 + l16];
    float ca0 = (cb0.z - cb0.x) * (cb0.w - cb0.y);
    float ca1 = (cb1.z - cb1.x) * (cb1.w - cb1.y);

    // A (16x4 f32): lanes 0-15 hold K=0 (VGPR0) and K=1 (VGPR1);
    // lanes 16-31 hold K=2,3 -> zero. A row m = [area_m, 1, 0, 0].
    v2f A;
    A.x = hi ? 0.0f : rowArea;
    A.y = hi ? 0.0f : 1.0f;
    // B (4x16 f32): VGPR0 = rows K=0 (lanes 0-15) / K=2 (lanes 16-31),
    // VGPR1 = rows K=1 / K=3. Rows: K0=ones, K1=area_col, K2=K3=0.
    v2f B0, B1;
    B0.x = hi ? 0.0f : 1.0f;  B0.y = hi ? 0.0f : ca0;
    B1.x = hi ? 0.0f : 1.0f;  B1.y = hi ? 0.0f : ca1;
    v8f C = {};

    // D[m][n] = area_m + area_n for the 16x16 sub-tile (matrix pipe).
    v8f D0 = __builtin_amdgcn_wmma_f32_16x16x4_f32(
        false, A, false, B0, (short)0, C, false, false);
    v8f D1 = __builtin_amdgcn_wmma_f32_16x16x4_f32(
        false, A, false, B1, (short)0, C, false, false);

    // Row boxes: accumulator r maps to row r + 8*(lane>=16).
    float4 rbox[8];
#pragma unroll
    for (int r = 0; r < 8; ++r)
        rbox[r] = sboxes[rowBase + r + (hi ? 8 : 0)];

    uint32_t wlo[8], whi[8];
#pragma unroll
    for (int r = 0; r < 8; ++r) {
        float4 rb = rbox[r];
        // sub-tile 0 (cols colBase .. colBase+15)
        float iw0 = fminf(rb.z, cb0.z) - fmaxf(rb.x, cb0.x);
        float ih0 = fminf(rb.w, cb0.w) - fmaxf(rb.y, cb0.y);
        float in0 = fmaxf(iw0, 0.0f) * fmaxf(ih0, 0.0f);
        bool p0 = in0 > IOU_T * (D0[r] - in0);      // iou > T, division-free
        // sub-tile 1 (cols colBase+16 .. colBase+31)
        float iw1 = fminf(rb.z, cb1.z) - fmaxf(rb.x, cb1.x);
        float ih1 = fminf(rb.w, cb1.w) - fmaxf(rb.y, cb1.y);
        float in1 = fmaxf(iw1, 0.0f) * fmaxf(ih1, 0.0f);
        bool p1 = in1 > IOU_T * (D1[r] - in1);

        uint32_t b0 = (uint32_t)__ballot(p0);
        uint32_t b1 = (uint32_t)__ballot(p1);
        wlo[r] = (b0 & 0xFFFFu) | ((b1 & 0xFFFFu) << 16);   // row r
        whi[r] = (b0 >> 16)     | (b1 & 0xFFFF0000u);       // row r+8
    }

    // Lane l (<16) writes the mask word for row rowBase+l.
    uint32_t w = 0;
#pragma unroll
    for (int r = 0; r < 8; ++r) {
        w = (lane == r)     ? wlo[r] : w;
        w = (lane == r + 8) ? whi[r] : w;
    }
    if (lane < 16)
        rowmask[(size_t)(rowBase + lane) * NWORDS + (colBase >> 5)] = w;
}

// ---------------------------------------------------------------------------
// Kernel 3: serial greedy scan. One block, 128 threads; removed[128] words in
// LDS, one word per thread. Bit i is only ever set by rows < i, so reading it
// in iteration i is race-free with one barrier per iteration.
//
// CDNA5 path: the next mask row is streamed into an LDS double buffer with
// GLOBAL_LOAD_ASYNC_TO_LDS_B32 (ASYNCcnt), issued right after the barrier and
// retired with s_wait_asynccnt 0 at the top of the next iteration — the L2
// round-trip overlaps one full iteration of processing instead of sitting on
// the serial dependency chain. Buffer (i+1)&1 is only rewritten after the
// barrier that proves every thread finished reading it (row i-1): no WAR race.
// ---------------------------------------------------------------------------
__global__ void nms_scan_kernel(const uint32_t* __restrict__ rowmask,
                                uint32_t* __restrict__ removed_out) {
    __shared__ uint32_t removed[NWORDS];
    const int t = threadIdx.x;
    removed[t] = 0;

#ifdef USE_ASYNC_LDS
    __shared__ uint32_t stage[2][NWORDS];
    // Prime: async-load row 0 into buffer 0 (one op in flight per thread).
    ASYNC_LOAD_B32(&rowmask[t], &stage[0][t]);
    for (int i = 0; i < NBOX; ++i) {
        WAIT_ASYNC0();              // this thread's word of row i has landed
        __syncthreads();            // publish row i + prev removed updates
        if (i + 1 < NBOX)           // safe: old readers of this buffer done
            ASYNC_LOAD_B32(&rowmask[(size_t)(i + 1) * NWORDS + t],
                           &stage[(i + 1) & 1][t]);
        uint32_t wi = removed[i >> 5];
        bool alive = ((wi >> (i & 31)) & 1u) == 0u;   // uniform across block
        if (alive) {
            uint32_t m = stage[i & 1][t];
            int iw = i >> 5;
            if (t < iw) {
                m = 0;
            } else if (t == iw) {
                int b = i & 31;
                uint32_t lowbits = (b == 31) ? 0xFFFFFFFFu
                                             : ((1u << (b + 1)) - 1u);
                m &= ~lowbits;                         // only suppress j > i
            }
            removed[t] |= m;
        }
    }
    __syncthreads();
#else
    __syncthreads();
    for (int i = 0; i < NBOX; ++i) {
        if (i + 1 < NBOX)
            __builtin_prefetch(&rowmask[(size_t)(i + 1) * NWORDS + t], 0, 0);
        uint32_t wi = removed[i >> 5];
        bool alive = ((wi >> (i & 31)) & 1u) == 0u;
        if (alive) {
            uint32_t m = rowmask[(size_t)i * NWORDS + t];
            int iw = i >> 5;
            if (t < iw) {
                m = 0;
            } else if (t == iw) {
                int b = i & 31;
                uint32_t lowbits = (b == 31) ? 0xFFFFFFFFu
                                             : ((1u << (b + 1)) - 1u);
                m &= ~lowbits;
            }
            removed[t] |= m;
        }
        __syncthreads();
    }
#endif
    removed_out[t] = removed[t];
}

// ---------------------------------------------------------------------------
// Kernel 4: masked output write: [x1 y1 x2 y2 score] in sorted order, zeros
// where suppressed or score < CONF_T.
// ---------------------------------------------------------------------------
__global__ void nms_out_kernel(const float4* __restrict__ sboxes,
                               const float* __restrict__ sscores,
                               const uint32_t* __restrict__ removed,
                               float* __restrict__ out) {
    int i = blockIdx.x * blockDim.x + threadIdx.x;
    if (i >= NBOX) return;
    uint32_t w = removed[i >> 5];
    float s = sscores[i];
    bool kept = (((w >> (i & 31)) & 1u) == 0u) && (s >= CONF_T);
    float4 b = sboxes[i];
    float* o = out + (size_t)i * 5;
    o[0] = kept ? b.x : 0.0f;
    o[1] = kept ? b.y : 0.0f;
    o[2] = kept ? b.z : 0.0f;
    o[3] = kept ? b.w : 0.0f;
    o[4] = kept ? s   : 0.0f;
}

// ---------------------------------------------------------------------------
extern "C" void kernel_launch(void* const* d_in, const int* in_sizes, int n_in,
                              void* d_out, int out_size, void* d_ws, size_t ws_size,
                              hipStream_t stream) {
    const float* xyxy = (const float*)d_in[0];   // [4096,4] f32
    const float* conf = (const float*)d_in[1];   // [4096]   f32
    float* out = (float*)d_out;                  // [4096,5] f32

    char* ws = (char*)d_ws;
    float4*   sboxes  = (float4*)(ws);                               // 64 KB
    float*    sscores = (float*)(ws + 65536);                        // 16 KB
    uint32_t* removed = (uint32_t*)(ws + 65536 + 16384);             // 512 B
    uint32_t* rowmask = (uint32_t*)(ws + 65536 + 16384 + 512);       // 2 MB

    nms_sort_kernel<<<1, 1024, 0, stream>>>(xyxy, conf, sboxes, sscores);
    nms_mask_kernel<<<4096, 256, 0, stream>>>(sboxes, rowmask);      // 32768 wave-tiles
    nms_scan_kernel<<<1, NWORDS, 0, stream>>>(rowmask, removed);
    nms_out_kernel<<<(NBOX + 255) / 256, 256, 0, stream>>>(sboxes, sscores, removed, out);
}